// Lineq2v2nano_42709154791757
// MI455X (gfx1250) — compile-verified
//
#include <hip/hip_runtime.h>

// Wave32 CDNA5 (gfx1250) implementation.
// out[b,i,j,f] = relu( in[b,i,j,:]·w0 + tbias[b,f] + r2[b,i,f] + r3[b,j,f] + (i==j)·d[b,i,f] )
// where tbias folds totsum·w1 + bias; d folds rowsum·w4 + totsum·w5 + diag_bias.

typedef __attribute__((ext_vector_type(2))) float v2f;
typedef __attribute__((ext_vector_type(8))) float v8f;

constexpr int   Bn = 32, Nn = 128, Ln = 16, Fn = 32;
constexpr float NAVG = 50.0f;

__global__ void zero_kernel(float* __restrict__ p, int n) {
    int i = blockIdx.x * blockDim.x + threadIdx.x;
    if (i < n) p[i] = 0.0f;
}

// One block per (b,i) row: rowsum[b,i,l] = sum_j in[b,i,j,l] / NAVG, and
// accumulate scaled rowsum into totsum[b,l] (totsum_true = stored / NAVG).
__global__ void rowsum_kernel(const float* __restrict__ in,
                              float* __restrict__ rowsum,
                              float* __restrict__ totsum) {
    int row = blockIdx.x;                 // b*N + i
    int b   = row >> 7;
    const float* base = in + (size_t)row * Nn * Ln;
    int t = threadIdx.x;                  // 256 threads
    int l = t & 15, part = t >> 4;        // 16 partial sums per l
    float acc = 0.0f;
#pragma unroll 4
    for (int j = part; j < Nn; j += 16) acc += base[j * Ln + l];
    __shared__ float sd[256];
    sd[t] = acc;
    __syncthreads();
    if (t < 16) {
        float s = 0.0f;
#pragma unroll
        for (int p = 0; p < 16; ++p) s += sd[l + 16 * p];
        s *= (1.0f / NAVG);
        rowsum[row * Ln + l] = s;
        atomicAdd(&totsum[b * Ln + l], s);
    }
}

// One wave per (b,i): project rowsum/totsum through w[:,1..5] into F-space.
__global__ void prep_kernel(const float* __restrict__ rowsum,
                            const float* __restrict__ totsum,
                            const float* __restrict__ w,
                            const float* __restrict__ bias,
                            const float* __restrict__ diag_bias,
                            float* __restrict__ r2, float* __restrict__ r3,
                            float* __restrict__ dv, float* __restrict__ tbias) {
    int row = blockIdx.x;                 // b*N + i
    int b = row >> 7, i = row & 127;
    int f = threadIdx.x;                  // 32 threads
    float a1 = 0, a2 = 0, a3 = 0, a4 = 0, a5 = 0;
#pragma unroll
    for (int l = 0; l < Ln; ++l) {
        float rs = rowsum[row * Ln + l];
        float ts = totsum[b * Ln + l] * (1.0f / NAVG);
        const float* wl = w + l * 6 * Fn;
        a1 += ts * wl[1 * Fn + f];
        a2 += rs * wl[2 * Fn + f];
        a3 += rs * wl[3 * Fn + f];
        a4 += rs * wl[4 * Fn + f];
        a5 += ts * wl[5 * Fn + f];
    }
    r2[row * Fn + f] = a2;
    r3[row * Fn + f] = a3;
    dv[row * Fn + f] = a4 + a5 + diag_bias[f];
    if (i == 0) tbias[b * Fn + f] = a1 + bias[f];
}

// Main GEMM+epilogue. 8 waves/block, one wave per (b,i) row, 8 tiles of 16(j)x32(F).
// A: 16x16 f32 tile of inputs, contracted via 4x V_WMMA_F32_16X16X4_F32 per F-half.
__global__ void __launch_bounds__(256) main_kernel(
    const float* __restrict__ in, const float* __restrict__ w,
    const float* __restrict__ r2, const float* __restrict__ r3,
    const float* __restrict__ dv, const float* __restrict__ tbias,
    float* __restrict__ out) {
    int lane = threadIdx.x & 31;
    int wave = threadIdx.x >> 5;
    int row  = blockIdx.x * 8 + wave;     // b*N + i
    int b = row >> 7, i = row & 127;
    int ln   = lane & 15;
    int hi   = lane >> 4;                 // lanes 16-31 hold K+2 / M+8 halves
    int koff = hi * 2;

    // B-matrix registers (w0 columns), per F-half h and K-step kk.
    // ISA B layout (4x16 f32): VGPR0 = row K=4kk+koff, VGPR1 = row K=4kk+koff+1; N = ln.
    v2f bm[2][4];
#pragma unroll
    for (int h = 0; h < 2; ++h)
#pragma unroll
        for (int kk = 0; kk < 4; ++kk) {
            int f = h * 16 + ln;
            bm[h][kk].x = w[(4 * kk + koff)     * 6 * Fn + f];
            bm[h][kk].y = w[(4 * kk + koff + 1) * 6 * Fn + f];
        }

    float basef[2], dval[2];
#pragma unroll
    for (int h = 0; h < 2; ++h) {
        int f = h * 16 + ln;
        basef[h] = tbias[b * Fn + f] + r2[row * Fn + f];
        dval[h]  = dv[row * Fn + f];
    }

    const float* rowbase = in  + (size_t)row * Nn * Ln;
    float*       outbase = out + (size_t)row * Nn * Fn;

    for (int j0 = 0; j0 < Nn; j0 += 16) {
        // A tile: pixel M = j0+ln; lane half picks K pair (koff).
        const float* ab = rowbase + (size_t)(j0 + ln) * Ln + koff;
        v2f a[4];
#pragma unroll
        for (int kk = 0; kk < 4; ++kk) {
            a[kk].x = ab[4 * kk];
            a[kk].y = ab[4 * kk + 1];
        }

        // Seed accumulators with broadcast terms: C/D layout M = v + hi*8, F = h*16+ln.
        v8f c0, c1;
#pragma unroll
        for (int v = 0; v < 8; ++v) {
            int j = j0 + v + hi * 8;
            float rr0 = r3[(b * Nn + j) * Fn + ln];
            float rr1 = r3[(b * Nn + j) * Fn + 16 + ln];
            float sel0 = (j == i) ? dval[0] : 0.0f;
            float sel1 = (j == i) ? dval[1] : 0.0f;
            c0[v] = basef[0] + rr0 + sel0;
            c1[v] = basef[1] + rr1 + sel1;
        }

#pragma unroll
        for (int kk = 0; kk < 4; ++kk) {
            c0 = __builtin_amdgcn_wmma_f32_16x16x4_f32(
                false, a[kk], false, bm[0][kk], (short)0, c0, false, false);
            c1 = __builtin_amdgcn_wmma_f32_16x16x4_f32(
                false, a[kk], false, bm[1][kk], (short)0, c1, false, false);
        }

#pragma unroll
        for (int v = 0; v < 8; ++v) {
            int j = j0 + v + hi * 8;
            float* op = outbase + (size_t)j * Fn;
            op[ln]      = fmaxf(c0[v], 0.0f);
            op[16 + ln] = fmaxf(c1[v], 0.0f);
        }
    }
}

extern "C" void kernel_launch(void* const* d_in, const int* in_sizes, int n_in,
                              void* d_out, int out_size, void* d_ws, size_t ws_size,
                              hipStream_t stream) {
    const float* in    = (const float*)d_in[0];  // [B,N,N,L]
    const float* w     = (const float*)d_in[1];  // [L,6,F]
    const float* bias  = (const float*)d_in[2];  // [F]
    const float* dbias = (const float*)d_in[3];  // [F]
    float* ws = (float*)d_ws;

    // Workspace layout (floats): total 460,288 floats (~1.84 MB)
    float* rowsum = ws;              // B*N*L  = 65536
    float* totsum = ws + 65536;      // B*L    = 512
    float* r2v    = ws + 66048;      // B*N*F  = 131072
    float* r3v    = ws + 197120;     // B*N*F  = 131072
    float* dvv    = ws + 328192;     // B*N*F  = 131072
    float* tbv    = ws + 459264;     // B*F    = 1024

    zero_kernel<<<2, 256, 0, stream>>>(totsum, Bn * Ln);
    rowsum_kernel<<<Bn * Nn, 256, 0, stream>>>(in, rowsum, totsum);
    prep_kernel<<<Bn * Nn, 32, 0, stream>>>(rowsum, totsum, w, bias, dbias,
                                            r2v, r3v, dvv, tbv);
    main_kernel<<<Bn * Nn / 8, 256, 0, stream>>>(in, w, r2v, r3v, dvv, tbv,
                                                 (float*)d_out);
}